// TLModel_2070174236838
// MI455X (gfx1250) — compile-verified
//
#include <hip/hip_runtime.h>
#include <hip/hip_bf16.h>

typedef __attribute__((ext_vector_type(2))) float v2f;
typedef __attribute__((ext_vector_type(8))) float v8f;

#define BB 32768
#define DD 1024
#define SS 64
#define OO 4
#define TILE_M 16
#define MAX_TILES ((BB + TILE_M * SS) / TILE_M)   // 2112 (every subject padded to 16)
#define SORTED_CAP (BB + TILE_M * SS)             // 33792

// Workspace layout (int32 elements)
#define OFF_COUNTS   0                      // 64
#define OFF_COUNTERS 64                     // 64
#define OFF_PADOFF   128                    // 65
#define OFF_TILESID  256                    // MAX_TILES
#define OFF_SORTED   (256 + MAX_TILES)      // SORTED_CAP
#define WS_INTS      (OFF_SORTED + SORTED_CAP)

// ---------------- pass 1: binning ----------------

__global__ void k_init(int* __restrict__ ws) {
  int i = blockIdx.x * blockDim.x + threadIdx.x;
  if (i < 128)        ws[OFF_COUNTS + i]  = 0;    // counts + counters
  if (i < MAX_TILES)  ws[OFF_TILESID + i] = -1;
  if (i < SORTED_CAP) ws[OFF_SORTED + i]  = -1;
}

__global__ void k_count(const int* __restrict__ sid, int* __restrict__ ws) {
  int i = blockIdx.x * blockDim.x + threadIdx.x;
  if (i < BB) atomicAdd(&ws[OFF_COUNTS + sid[i]], 1);
}

// one block of 64 threads: padded (mul-of-16) prefix offsets + per-tile subject id
__global__ void k_scan(int* __restrict__ ws) {
  __shared__ int sh_off[SS + 1];
  if (threadIdx.x == 0) {
    int acc = 0;
    for (int s = 0; s < SS; ++s) {
      sh_off[s] = acc;
      acc += (ws[OFF_COUNTS + s] + (TILE_M - 1)) & ~(TILE_M - 1);
    }
    sh_off[SS] = acc;
  }
  __syncthreads();
  int s = threadIdx.x;   // 0..63
  ws[OFF_PADOFF + s] = sh_off[s];
  if (s == 0) ws[OFF_PADOFF + SS] = sh_off[SS];
  int t0 = sh_off[s] >> 4;
  int t1 = sh_off[s + 1] >> 4;
  for (int t = t0; t < t1; ++t) ws[OFF_TILESID + t] = s;
}

__global__ void k_scatter(const int* __restrict__ sid, int* __restrict__ ws) {
  int i = blockIdx.x * blockDim.x + threadIdx.x;
  if (i >= BB) return;
  int s = sid[i];
  int pos = ws[OFF_PADOFF + s] + atomicAdd(&ws[OFF_COUNTERS + s], 1);
  ws[OFF_SORTED + pos] = i;
}

// ---------------- pass 2: WMMA tiles ----------------
// One wave (32 lanes) per 16-sample tile; all samples in a tile share a subject.
// A: x rows (16x4 f32 per step), B: W[s] columns replicated mod 4 (4x16 f32),
// C: f32 16x16 accumulator; 256 x v_wmma_f32_16x16x4_f32 per tile.

__global__ void __launch_bounds__(256) k_gemm(
    const float* __restrict__ x, const float* __restrict__ W,
    const float* __restrict__ bias, const int* __restrict__ ws,
    float* __restrict__ out) {
  const int* tile_sid = ws + OFF_TILESID;
  const int* sorted   = ws + OFF_SORTED;

  int tile = (blockIdx.x * blockDim.x + threadIdx.x) >> 5;   // one wave per tile
  if (tile >= MAX_TILES) return;                             // wave-uniform exit
  int s = tile_sid[tile];
  if (s < 0) return;                                         // wave-uniform exit

  int lane = threadIdx.x & 31;
  int m  = lane & 15;     // A row / D column
  int h  = lane >> 4;     // lane half -> K offset +2, D row offset +8
  int n3 = lane & 3;      // replicated output column for B

  int idx = sorted[tile * TILE_M + m];
  const float2* xp = (const float2*)(x + (size_t)(idx < 0 ? 0 : idx) * DD);
  const float*  wp = W + (size_t)s * (DD * OO) + n3;

  v8f c = {};
  for (int k = 0; k < DD; k += 16) {
#pragma unroll
    for (int j = 0; j < 4; ++j) {
      int kb = k + 4 * j;
      // A lane (m,h): x[idx[m], kb+2h .. kb+2h+1]  -> one b64 load
      float2 av = xp[(kb >> 1) + h];
      v2f a; a.x = av.x; a.y = av.y;
      // B lane (n,h): W[s, kb+2h, n&3], W[s, kb+2h+1, n&3]
      v2f b; b.x = wp[(kb + 2 * h) * OO];
             b.y = wp[(kb + 2 * h + 1) * OO];
      c = __builtin_amdgcn_wmma_f32_16x16x4_f32(
              /*neg_a=*/false, a, /*neg_b=*/false, b,
              /*c_mod=*/(short)0, c, /*reuse_a=*/false, /*reuse_b=*/false);
    }
  }

  // C layout: VGPR r = rows {r, r+8h}; lane n = column. Only n<4 is real.
  int n = lane & 15;
  if (n < OO) {
    float bs = bias[s * OO + n];
#pragma unroll
    for (int r = 0; r < 8; ++r) {
      int mr = (h << 3) + r;
      int si = sorted[tile * TILE_M + mr];
      if (si >= 0) out[(size_t)si * OO + n] = c[r] + bs;
    }
  }
}

// ---------------- launcher ----------------

extern "C" void kernel_launch(void* const* d_in, const int* in_sizes, int n_in,
                              void* d_out, int out_size, void* d_ws, size_t ws_size,
                              hipStream_t stream) {
  const float* x    = (const float*)d_in[0];
  const int*   sid  = (const int*)d_in[1];
  const float* W    = (const float*)d_in[2];
  const float* bias = (const float*)d_in[3];
  float* out = (float*)d_out;
  int*   ws  = (int*)d_ws;
  (void)in_sizes; (void)n_in; (void)out_size; (void)ws_size;

  k_init   <<<(SORTED_CAP + 255) / 256, 256, 0, stream>>>(ws);
  k_count  <<<(BB + 255) / 256,        256, 0, stream>>>(sid, ws);
  k_scan   <<<1,                        64, 0, stream>>>(ws);
  k_scatter<<<(BB + 255) / 256,        256, 0, stream>>>(sid, ws);
  k_gemm   <<<(MAX_TILES * 32 + 255) / 256, 256, 0, stream>>>(x, W, bias, ws, out);
}